// RmseLoss_7301444403961
// MI455X (gfx1250) — compile-verified
//
#include <hip/hip_runtime.h>
#include <hip/hip_bf16.h>

// RMSE loss: masked column reduction over NT, then tiny epilogue.
// Memory-bound: 524 MB @ 23.3 TB/s => ~22.5 us floor. fp32 throughout.
// Reduction-over-t done with V_WMMA_F32_16X16X4_F32 (A = ones) so the
// accumulate rides the matrix pipe while loads stream from HBM.

#define NT    4096
#define NCOL  16000            // NS*NY = 4000*4
#define TCHUNKS 16             // gridDim.y
#define ROWS_PER_CHUNK (NT / TCHUNKS)   // 256 rows, 64 wmma steps

typedef __attribute__((ext_vector_type(2))) float v2f;
typedef __attribute__((ext_vector_type(8))) float v8f;

__global__ __launch_bounds__(256) void rmse_phase1(
    const float* __restrict__ outp, const float* __restrict__ tgtp,
    float* __restrict__ ws_ssum, float* __restrict__ ws_cnt)
{
    const int lane = threadIdx.x & 31;
    const int wave = threadIdx.x >> 5;
    const int j    = lane & 15;          // output column within wave tile
    const int half = lane >> 4;          // which row-pair this lane feeds
    const int col  = (blockIdx.x * 8 + wave) * 16 + j;   // 0..15999
    const int t0   = blockIdx.y * ROWS_PER_CHUNK;

    // A = all-ones 16x4: D[i][j] = sum_k B[k][j] (pure K-reduction)
    v2f a; a[0] = 1.0f; a[1] = 1.0f;
    v8f c_sq  = {};
    v8f c_cnt = {};

    const float* po = outp + (size_t)(t0 + 2 * half) * NCOL + col;
    const float* pt = tgtp + (size_t)(t0 + 2 * half) * NCOL + col;

    #pragma unroll 4
    for (int t = 0; t < ROWS_PER_CHUNK; t += 4) {
        // lane pair {j, j+16} covers rows t..t+3 of memory column `col`
        float o0 = po[0];      float g0 = pt[0];
        float o1 = po[NCOL];   float g1 = pt[NCOL];
        po += 4 * NCOL;        pt += 4 * NCOL;

        bool  m0 = !__builtin_isnan(g0);
        float d0 = o0 - g0;
        float s0 = m0 ? d0 * d0 : 0.0f;
        float f0 = m0 ? 1.0f : 0.0f;

        bool  m1 = !__builtin_isnan(g1);
        float d1 = o1 - g1;
        float s1 = m1 ? d1 * d1 : 0.0f;
        float f1 = m1 ? 1.0f : 0.0f;

        v2f bsq; bsq[0] = s0; bsq[1] = s1;
        v2f bm;  bm[0]  = f0; bm[1]  = f1;

        // (neg_a, A, neg_b, B, c_mod, C, reuse_a, reuse_b)
        c_sq  = __builtin_amdgcn_wmma_f32_16x16x4_f32(
                    false, a, false, bsq, (short)0, c_sq,  false, false);
        c_cnt = __builtin_amdgcn_wmma_f32_16x16x4_f32(
                    false, a, false, bm,  (short)0, c_cnt, false, false);
    }

    // All D rows are identical (A is ones); c[0] on lanes 0..15 holds the
    // per-column sums for N = lane. One atomic per column per block.
    if (lane < 16) {
        atomicAdd(&ws_ssum[col], c_sq[0]);
        atomicAdd(&ws_cnt[col],  c_cnt[0]);
    }
}

__global__ __launch_bounds__(256) void rmse_phase2(
    const float* __restrict__ ws_ssum, const float* __restrict__ ws_cnt,
    float* __restrict__ d_out)
{
    __shared__ float s_r[256];
    __shared__ float s_n[256];

    float rsum = 0.0f, nv = 0.0f;
    // col % 4 == threadIdx.x % 4, so each thread's partial belongs to one y.
    for (int col = threadIdx.x; col < NCOL; col += 256) {
        float ss = ws_ssum[col];
        float cn = ws_cnt[col];
        bool valid = cn > 0.0f;
        float mse = ss / fmaxf(cn, 1.0f);
        rsum += valid ? sqrtf(mse) : 0.0f;
        nv   += valid ? 1.0f : 0.0f;
    }
    s_r[threadIdx.x] = rsum;
    s_n[threadIdx.x] = nv;
    __syncthreads();

    if (threadIdx.x < 4) {
        int y = threadIdx.x;
        float R = 0.0f, N = 0.0f;
        for (int i = y; i < 256; i += 4) { R += s_r[i]; N += s_n[i]; }
        s_r[y] = R / N;                 // per-variable mean rmse
    }
    __syncthreads();

    if (threadIdx.x == 0)
        d_out[0] = 0.25f * (s_r[0] + s_r[1] + s_r[2] + s_r[3]);
}

__global__ void rmse_init(float* __restrict__ ws, int n)
{
    int i = blockIdx.x * blockDim.x + threadIdx.x;
    if (i < n) ws[i] = 0.0f;
}

extern "C" void kernel_launch(void* const* d_in, const int* in_sizes, int n_in,
                              void* d_out, int out_size, void* d_ws, size_t ws_size,
                              hipStream_t stream)
{
    const float* outp = (const float*)d_in[0];   // output [NT, NS, NY] f32
    const float* tgtp = (const float*)d_in[1];   // target [NT, NS, NY] f32 (with NaNs)

    float* ws_ssum = (float*)d_ws;               // [NCOL]
    float* ws_cnt  = ws_ssum + NCOL;             // [NCOL]

    rmse_init<<<(2 * NCOL + 255) / 256, 256, 0, stream>>>(ws_ssum, 2 * NCOL);

    dim3 grid(NCOL / 128, TCHUNKS);              // 125 x 16 blocks
    rmse_phase1<<<grid, 256, 0, stream>>>(outp, tgtp, ws_ssum, ws_cnt);

    rmse_phase2<<<1, 256, 0, stream>>>(ws_ssum, ws_cnt, (float*)d_out);
}